// ResNetBlock_41798621725224
// MI455X (gfx1250) — compile-verified
//
#include <hip/hip_runtime.h>

// ---------------------------------------------------------------------------
// ResNet sparse-conv block on MI455X (gfx1250), fp32 via V_WMMA_F32_16X16X4_F32
//   h1 = relu(BN(x @ W1))                 [N,256]x[256,64] -> [N,64]
//   h2 = relu(BN(einsum(h1[nbr], W3)))    27x gather, [64,64] mats  (dominant)
//   o  = relu(BN(h2 @ W2) + x)            [N,64]x[64,256] -> [N,256]
// Each wave owns a 32-point x 64-col tile; weights pre-packed to fragment
// order so every B fragment is one coalesced b64 load per lane.
// ---------------------------------------------------------------------------

typedef __attribute__((ext_vector_type(2))) float v2f;
typedef __attribute__((ext_vector_type(8))) float v8f;

__device__ __forceinline__ v8f wmma_f32(v2f a, v2f b, v8f c) {
  return __builtin_amdgcn_wmma_f32_16x16x4_f32(false, a, false, b,
                                               (short)0, c, false, false);
}

#define INC  256
#define MIDC 64
#define OUTC 256
#define KNBR 27

// ---------------------------------------------------------------------------
// Zero a small float region (stats accumulators)
// ---------------------------------------------------------------------------
__global__ void zero_kernel(float* __restrict__ p, int n) {
  int i = blockIdx.x * blockDim.x + threadIdx.x;
  if (i < n) p[i] = 0.0f;
}

// ---------------------------------------------------------------------------
// Pack weights into WMMA B-fragment order.
//   Input : W [B][K][Nout] row-major
//   Output: per batch, [S=K/4][NC=Nout/16][lane(32)][2]  (each fragment is a
//           contiguous 256B block; lane L loads bytes L*8..L*8+8 as one b64)
// Fragment element (lane,t): row = s*4 + (lane>=16 ? 2 : 0) + t,
//                            col = nc*16 + (lane&15)
// ---------------------------------------------------------------------------
__global__ void pack_weights(const float* __restrict__ W, float* __restrict__ P,
                             int B, int K, int Nout) {
  int total = B * K * Nout;
  int e = blockIdx.x * blockDim.x + threadIdx.x;
  if (e >= total) return;
  int per  = K * Nout;
  int b    = e / per;
  int r    = e - b * per;
  int t    = r & 1;
  int lane = (r >> 1) & 31;
  int NC   = Nout >> 4;
  int nc   = (r >> 6) % NC;
  int s    = r / (64 * NC);
  int row  = s * 4 + (lane >> 4) * 2 + t;
  int col  = nc * 16 + (lane & 15);
  P[e] = W[(size_t)b * per + (size_t)row * Nout + col];
}

// ---------------------------------------------------------------------------
// GEMM1: x[N,256] @ W1 -> out[N,64].  4 waves/block, 32 points per wave.
// ---------------------------------------------------------------------------
__global__ __launch_bounds__(128) void gemm_x_w1(const float* __restrict__ x,
                                                 const float* __restrict__ WP,
                                                 float* __restrict__ out, int N) {
  const int lane = threadIdx.x & 31;
  const int wave = threadIdx.x >> 5;
  const int m0   = (blockIdx.x * 4 + wave) * 32;
  if (m0 >= N) return;
  const int r    = lane & 15;
  const int half = lane >> 4;

  int p0 = m0 + r;      if (p0 >= N) p0 = N - 1;
  int p1 = m0 + 16 + r; if (p1 >= N) p1 = N - 1;
  const float* x0 = x + (size_t)p0 * INC;
  const float* x1 = x + (size_t)p1 * INC;

  v8f acc[2][4] = {};

  for (int s = 0; s < INC / 4; ++s) {
    const int k0 = s * 4 + half * 2;
    v2f a0 = *(const v2f*)(x0 + k0);
    v2f a1 = *(const v2f*)(x1 + k0);
    const float* F = WP + (size_t)s * 256 + lane * 2;   // 4 fragments of 64
#pragma unroll
    for (int nc = 0; nc < 4; ++nc) {
      v2f b = *(const v2f*)(F + nc * 64);
      acc[0][nc] = wmma_f32(a0, b, acc[0][nc]);
      acc[1][nc] = wmma_f32(a1, b, acc[1][nc]);
    }
  }

  if (m0 + 32 <= N) {
#pragma unroll
    for (int t = 0; t < 2; ++t)
#pragma unroll
      for (int nc = 0; nc < 4; ++nc)
#pragma unroll
        for (int j = 0; j < 8; ++j)
          out[(size_t)(m0 + t * 16 + half * 8 + j) * MIDC + nc * 16 + r] =
              acc[t][nc][j];
  } else {
#pragma unroll
    for (int t = 0; t < 2; ++t)
#pragma unroll
      for (int nc = 0; nc < 4; ++nc)
#pragma unroll
        for (int j = 0; j < 8; ++j) {
          int m = m0 + t * 16 + half * 8 + j;
          if (m < N) out[(size_t)m * MIDC + nc * 16 + r] = acc[t][nc][j];
        }
  }
}

// ---------------------------------------------------------------------------
// Gather GEMM: out[n,:] = sum_k h1[nbr[n,k],:] @ W3[k]   (dominant kernel)
// h1 (30.7MB) and packed W3 (442KB) are L2-resident.
// ---------------------------------------------------------------------------
__global__ __launch_bounds__(128) void gemm_gather_w3(const float* __restrict__ h1,
                                                      const int* __restrict__ nbr,
                                                      const float* __restrict__ WP3,
                                                      float* __restrict__ out, int N) {
  const int lane = threadIdx.x & 31;
  const int wave = threadIdx.x >> 5;
  const int m0   = (blockIdx.x * 4 + wave) * 32;
  if (m0 >= N) return;
  const int r    = lane & 15;
  const int half = lane >> 4;

  int p0 = m0 + r;      if (p0 >= N) p0 = N - 1;
  int p1 = m0 + 16 + r; if (p1 >= N) p1 = N - 1;

  v8f acc[2][4] = {};

  for (int k = 0; k < KNBR; ++k) {
    const int row0 = nbr[(size_t)p0 * KNBR + k];
    const int row1 = nbr[(size_t)p1 * KNBR + k];
    const float* a0r = h1 + (size_t)row0 * MIDC;
    const float* a1r = h1 + (size_t)row1 * MIDC;
    const float* Wk  = WP3 + (size_t)k * (MIDC * MIDC);
#pragma unroll
    for (int s = 0; s < MIDC / 4; ++s) {
      const int k0 = s * 4 + half * 2;
      v2f a0 = *(const v2f*)(a0r + k0);
      v2f a1 = *(const v2f*)(a1r + k0);
      const float* F = Wk + (size_t)s * 256 + lane * 2;
#pragma unroll
      for (int nc = 0; nc < 4; ++nc) {
        v2f b = *(const v2f*)(F + nc * 64);
        acc[0][nc] = wmma_f32(a0, b, acc[0][nc]);
        acc[1][nc] = wmma_f32(a1, b, acc[1][nc]);
      }
    }
  }

  if (m0 + 32 <= N) {
#pragma unroll
    for (int t = 0; t < 2; ++t)
#pragma unroll
      for (int nc = 0; nc < 4; ++nc)
#pragma unroll
        for (int j = 0; j < 8; ++j)
          out[(size_t)(m0 + t * 16 + half * 8 + j) * MIDC + nc * 16 + r] =
              acc[t][nc][j];
  } else {
#pragma unroll
    for (int t = 0; t < 2; ++t)
#pragma unroll
      for (int nc = 0; nc < 4; ++nc)
#pragma unroll
        for (int j = 0; j < 8; ++j) {
          int m = m0 + t * 16 + half * 8 + j;
          if (m < N) out[(size_t)m * MIDC + nc * 16 + r] = acc[t][nc][j];
        }
  }
}

// ---------------------------------------------------------------------------
// GEMM2: h2[N,64] @ W2[64,256] -> out[N,256].
// One block = one 32-point tile; wave w covers cols [w*64, w*64+64).
// ---------------------------------------------------------------------------
__global__ __launch_bounds__(128) void gemm_w2(const float* __restrict__ h2,
                                               const float* __restrict__ WP2,
                                               float* __restrict__ out, int N) {
  const int lane = threadIdx.x & 31;
  const int wave = threadIdx.x >> 5;
  const int m0   = blockIdx.x * 32;
  if (m0 >= N) return;
  const int r    = lane & 15;
  const int half = lane >> 4;

  int p0 = m0 + r;      if (p0 >= N) p0 = N - 1;
  int p1 = m0 + 16 + r; if (p1 >= N) p1 = N - 1;
  const float* a0r = h2 + (size_t)p0 * MIDC;
  const float* a1r = h2 + (size_t)p1 * MIDC;

  v8f acc[2][4] = {};

#pragma unroll
  for (int s = 0; s < MIDC / 4; ++s) {
    const int k0 = s * 4 + half * 2;
    v2f a0 = *(const v2f*)(a0r + k0);
    v2f a1 = *(const v2f*)(a1r + k0);
    // packed layout: [s][16 nc][32 lanes][2]; this wave uses nc = wave*4..+3
    const float* F = WP2 + (((size_t)s * 16 + wave * 4) * 32 + lane) * 2;
#pragma unroll
    for (int nc = 0; nc < 4; ++nc) {
      v2f b = *(const v2f*)(F + nc * 64);
      acc[0][nc] = wmma_f32(a0, b, acc[0][nc]);
      acc[1][nc] = wmma_f32(a1, b, acc[1][nc]);
    }
  }

  if (m0 + 32 <= N) {
#pragma unroll
    for (int t = 0; t < 2; ++t)
#pragma unroll
      for (int nc = 0; nc < 4; ++nc)
#pragma unroll
        for (int j = 0; j < 8; ++j)
          out[(size_t)(m0 + t * 16 + half * 8 + j) * OUTC + wave * 64 +
              nc * 16 + r] = acc[t][nc][j];
  } else {
#pragma unroll
    for (int t = 0; t < 2; ++t)
#pragma unroll
      for (int nc = 0; nc < 4; ++nc)
#pragma unroll
        for (int j = 0; j < 8; ++j) {
          int m = m0 + t * 16 + half * 8 + j;
          if (m < N)
            out[(size_t)m * OUTC + wave * 64 + nc * 16 + r] = acc[t][nc][j];
        }
  }
}

// ---------------------------------------------------------------------------
// Column statistics: per-channel sum / sum-of-squares over N rows.
// ---------------------------------------------------------------------------
__global__ void col_stats(const float* __restrict__ src, int N, int C,
                          float* __restrict__ sum, float* __restrict__ sumsq) {
  const int c = threadIdx.x;
  float s = 0.0f, q = 0.0f;
  for (int n = blockIdx.x; n < N; n += gridDim.x) {
    float v = src[(size_t)n * C + c];
    s += v; q += v * v;
  }
  atomicAdd(&sum[c], s);
  atomicAdd(&sumsq[c], q);
}

// ---------------------------------------------------------------------------
__global__ void bn_finalize(const float* __restrict__ sum, const float* __restrict__ sumsq,
                            const float* __restrict__ g, const float* __restrict__ b,
                            float* __restrict__ scale, float* __restrict__ shift,
                            int C, float invN) {
  int c = blockIdx.x * blockDim.x + threadIdx.x;
  if (c < C) {
    float m  = sum[c] * invN;
    float v  = sumsq[c] * invN - m * m;
    float rs = rsqrtf(v + 1e-5f);
    float sc = g[c] * rs;
    scale[c] = sc;
    shift[c] = b[c] - m * sc;
  }
}

// ---------------------------------------------------------------------------
__global__ void bn_relu_apply(const float* __restrict__ in, float* __restrict__ out,
                              const float* __restrict__ scale,
                              const float* __restrict__ shift,
                              int total, int Cmask) {
  int i = blockIdx.x * blockDim.x + threadIdx.x;
  if (i < total) {
    int c = i & Cmask;
    float v = in[i] * scale[c] + shift[c];
    out[i] = v > 0.0f ? v : 0.0f;
  }
}

// ---------------------------------------------------------------------------
__global__ void final_residual(const float* __restrict__ pre, const float* __restrict__ x,
                               const float* __restrict__ scale,
                               const float* __restrict__ shift,
                               float* __restrict__ out, int total) {
  int i = blockIdx.x * blockDim.x + threadIdx.x;
  if (i < total) {
    int c = i & (OUTC - 1);
    float v = pre[i] * scale[c] + shift[c] + x[i];
    out[i] = v > 0.0f ? v : 0.0f;
  }
}

// ---------------------------------------------------------------------------
extern "C" void kernel_launch(void* const* d_in, const int* in_sizes, int n_in,
                              void* d_out, int out_size, void* d_ws, size_t ws_size,
                              hipStream_t stream) {
  const float* x   = (const float*)d_in[0];
  const int*   nbr = (const int*)  d_in[1];
  const float* W1  = (const float*)d_in[2];
  const float* W3  = (const float*)d_in[3];
  const float* W2  = (const float*)d_in[4];
  const float* g1  = (const float*)d_in[5];
  const float* b1  = (const float*)d_in[6];
  const float* g2  = (const float*)d_in[7];
  const float* b2  = (const float*)d_in[8];
  const float* g3  = (const float*)d_in[9];
  const float* b3  = (const float*)d_in[10];
  float* out = (float*)d_out;

  const int N = in_sizes[1] / KNBR;   // nbr_idx is [N, 27]

  // Workspace layout (floats)
  float* ws  = (float*)d_ws;
  float* H1  = ws;                         // [N,64]
  float* H2  = H1 + (size_t)N * MIDC;      // [N,64]
  float* OP  = H2 + (size_t)N * MIDC;      // [N,256]
  float* ST  = OP + (size_t)N * OUTC;      // stats (1536 floats)
  float* WP1 = ST + 1536;                  // packed W1: 256*64
  float* WP3 = WP1 + INC * MIDC;           // packed W3: 27*64*64
  float* WP2 = WP3 + KNBR * MIDC * MIDC;   // packed W2: 64*256
  float* sum1 = ST;        float* sq1 = ST + 64;   float* sc1 = ST + 128;  float* sh1 = ST + 192;
  float* sum2 = ST + 256;  float* sq2 = ST + 320;  float* sc2 = ST + 384;  float* sh2 = ST + 448;
  float* sum3 = ST + 512;  float* sq3 = ST + 768;  float* sc3 = ST + 1024; float* sh3 = ST + 1280;

  const float invN = 1.0f / (float)N;
  const int tiles  = (N + 31) / 32;
  const int totalM = N * MIDC;
  const int totalO = N * OUTC;

  zero_kernel<<<(1536 + 255) / 256, 256, 0, stream>>>(ST, 1536);
  pack_weights<<<(INC * MIDC + 255) / 256, 256, 0, stream>>>(W1, WP1, 1, INC, MIDC);
  pack_weights<<<(KNBR * MIDC * MIDC + 255) / 256, 256, 0, stream>>>(W3, WP3, KNBR, MIDC, MIDC);
  pack_weights<<<(MIDC * OUTC + 255) / 256, 256, 0, stream>>>(W2, WP2, 1, MIDC, OUTC);

  // Stage 1: h1 = relu(BN1(x @ W1))
  gemm_x_w1<<<(tiles + 3) / 4, 128, 0, stream>>>(x, WP1, H1, N);
  col_stats<<<512, MIDC, 0, stream>>>(H1, N, MIDC, sum1, sq1);
  bn_finalize<<<1, MIDC, 0, stream>>>(sum1, sq1, g1, b1, sc1, sh1, MIDC, invN);
  bn_relu_apply<<<(totalM + 255) / 256, 256, 0, stream>>>(H1, H1, sc1, sh1, totalM, MIDC - 1);

  // Stage 2: h2 = relu(BN2(sum_k h1[nbr] @ W3[k]))
  gemm_gather_w3<<<(tiles + 3) / 4, 128, 0, stream>>>(H1, nbr, WP3, H2, N);
  col_stats<<<512, MIDC, 0, stream>>>(H2, N, MIDC, sum2, sq2);
  bn_finalize<<<1, MIDC, 0, stream>>>(sum2, sq2, g2, b2, sc2, sh2, MIDC, invN);
  bn_relu_apply<<<(totalM + 255) / 256, 256, 0, stream>>>(H2, H2, sc2, sh2, totalM, MIDC - 1);

  // Stage 3: out = relu(BN3(h2 @ W2) + x)
  gemm_w2<<<tiles, 128, 0, stream>>>(H2, WP2, OP, N);
  col_stats<<<512, OUTC, 0, stream>>>(OP, N, OUTC, sum3, sq3);
  bn_finalize<<<1, OUTC, 0, stream>>>(sum3, sq3, g3, b3, sc3, sh3, OUTC, invN);
  final_residual<<<(totalO + 255) / 256, 256, 0, stream>>>(OP, x, sc3, sh3, out, totalO);
}